// WeightedRankNet_36687610643030
// MI455X (gfx1250) — compile-verified
//
#include <hip/hip_runtime.h>
#include <math.h>

#define N_FEAT        136
#define COL_DOCLEN    14
#define COL_WHOLE_LEN 16
#define COL_TF        24
#define COL_INLINK    127
#define COL_OUTLINK   128
#define COL_PAGERANK  129

#define RED_BLOCK 256
#define TILE      4096   // rows of the column DMA'd to LDS per block (16 KB)

typedef unsigned int u32x4 __attribute__((ext_vector_type(4)));
typedef unsigned int u32x8 __attribute__((ext_vector_type(8)));

// ---------------------------------------------------------------------------
// Kernel 1: partial sums of global_features[:, COL_WHOLE_LEN] using the
// CDNA5 Tensor Data Mover. One TENSOR_LOAD_TO_LDS per block pulls a
// TILE x 1 tile of the (stride-136) column into LDS; rows past
// tensor_dim1 = n_docs - r0 are zero-filled by TDM OOB handling (free tail).
// ---------------------------------------------------------------------------
__global__ void col_mean_partial(const float* __restrict__ gf, int n_docs,
                                 float* __restrict__ partials) {
    __shared__ float lbuf[TILE];
    const unsigned tid = threadIdx.x;

    const unsigned r0    = blockIdx.x * (unsigned)TILE;
    const unsigned tdim1 = (unsigned)n_docs - r0;   // rows remaining (>0 by grid)

    if (tid < 32) {  // wave 0 issues the DMA (EXEC all-ones within the wave)
        // LDS byte offset of lbuf (low 32 bits of flat addr = WG-relative)
        const unsigned lds_base = (unsigned)(unsigned long long)&lbuf[0];
        const unsigned long long ga =
            (unsigned long long)(const void*)(gf + (size_t)r0 * N_FEAT + COL_WHOLE_LEN);

        // ---- D# group 0 (128b): count=1 | lds_addr | global_addr | type=2
        u32x4 g0;
        g0.x = 1u;                                           // count=1, user D#
        g0.y = lds_base;                                     // lds_addr (bytes)
        g0.z = (unsigned)(ga & 0xFFFFFFFFull);               // global_addr[31:0]
        g0.w = (unsigned)((ga >> 32) & 0x01FFFFFFull)        // global_addr[56:32]
             | (2u << 30);                                   // type=2 ("image")

        // ---- D# group 1 (256b)
        u32x8 g1;
        g1.s0 = 2u << 16;                    // workgroup_mask=0, data_size=2 (4B)
        g1.s1 = 1u << 16;                    // tensor_dim0[15:0] = 1
        g1.s2 = (tdim1 & 0xFFFFu) << 16;     // tensor_dim0 hi=0 | tensor_dim1 lo
        g1.s3 = (tdim1 >> 16)                // tensor_dim1 hi
              | (1u << 16);                  // tile_dim0 = 1
        g1.s4 = (unsigned)TILE;              // tile_dim1 = TILE, tile_dim2 = 0
        g1.s5 = (unsigned)N_FEAT;            // tensor_dim0_stride = 136 elements
        g1.s6 = 0u;                          // stride hi | tensor_dim1_stride lo
        g1.s7 = 0u;                          // tensor_dim1_stride hi

        // 2D tensor -> 2 SGPR-group form (VADDR2/VADDR3 = NULL)
        asm volatile("tensor_load_to_lds %0, %1" :: "s"(g0), "s"(g1) : "memory");
        __builtin_amdgcn_s_wait_tensorcnt(0);
    }
    __syncthreads();

    // Deterministic per-thread strided sum over the LDS tile, then tree reduce.
    float s = 0.0f;
    #pragma unroll
    for (unsigned k = 0; k < TILE / RED_BLOCK; ++k)
        s += lbuf[tid + k * RED_BLOCK];
    __syncthreads();
    lbuf[tid] = s;
    __syncthreads();
    for (unsigned off = RED_BLOCK / 2; off > 0; off >>= 1) {
        if (tid < off) lbuf[tid] += lbuf[tid + off];
        __syncthreads();
    }
    if (tid == 0) partials[blockIdx.x] = lbuf[0];
}

// ---------------------------------------------------------------------------
// Kernel 2: reduce partials -> avg_doc_len (deterministic fixed-order tree).
// ---------------------------------------------------------------------------
__global__ void col_mean_final(const float* __restrict__ partials, int nblk,
                               float* __restrict__ avg_out, int n_docs) {
    __shared__ float lbuf[RED_BLOCK];
    const unsigned tid = threadIdx.x;
    float s = 0.0f;
    for (unsigned i = tid; i < (unsigned)nblk; i += RED_BLOCK)
        s += partials[i];
    lbuf[tid] = s;
    __syncthreads();
    for (unsigned off = RED_BLOCK / 2; off > 0; off >>= 1) {
        if (tid < off) lbuf[tid] += lbuf[tid + off];
        __syncthreads();
    }
    if (tid == 0) *avg_out = lbuf[0] / (float)n_docs;
}

// ---------------------------------------------------------------------------
// Threefry-2x32, key = jax.random.key(1) => (k0,k1) = (0,1). 20 rounds.
// ---------------------------------------------------------------------------
__device__ __forceinline__ unsigned rotl32(unsigned x, unsigned n) {
    return (x << n) | (x >> (32u - n));
}

__device__ __forceinline__ void threefry2x32_key01(unsigned c0, unsigned c1,
                                                   unsigned& o0, unsigned& o1) {
    const unsigned ks0 = 0u;
    const unsigned ks1 = 1u;
    const unsigned ks2 = 0x1BD11BDAu ^ ks0 ^ ks1;  // 0x1BD11BDB
    unsigned x0 = c0 + ks0;
    unsigned x1 = c1 + ks1;
#define TF_R(r) { x0 += x1; x1 = rotl32(x1, (r)); x1 ^= x0; }
    TF_R(13) TF_R(15) TF_R(26) TF_R(6)
    x0 += ks1; x1 += ks2 + 1u;
    TF_R(17) TF_R(29) TF_R(16) TF_R(24)
    x0 += ks2; x1 += ks0 + 2u;
    TF_R(13) TF_R(15) TF_R(26) TF_R(6)
    x0 += ks0; x1 += ks1 + 3u;
    TF_R(17) TF_R(29) TF_R(16) TF_R(24)
    x0 += ks1; x1 += ks2 + 4u;
    TF_R(13) TF_R(15) TF_R(26) TF_R(6)
    x0 += ks2; x1 += ks0 + 5u;
#undef TF_R
    o0 = x0; o1 = x1;
}

// ---------------------------------------------------------------------------
// Kernel 3: gather + fused BM25 / link / freshness scoring.
// ---------------------------------------------------------------------------
__global__ void rank_score(const int* __restrict__ idxs,
                           const float* __restrict__ gf,
                           const float* __restrict__ p_k1,
                           const float* __restrict__ p_b,
                           const float* __restrict__ p_w,
                           const float* __restrict__ p_pr,
                           const float* __restrict__ p_il,
                           const float* __restrict__ p_ol,
                           const float* __restrict__ p_fr,
                           const float* __restrict__ avg_ptr,
                           float idf,
                           float* __restrict__ out, int n) {
    const int i = blockIdx.x * blockDim.x + threadIdx.x;
    if (i >= n) return;

    const float k1  = *p_k1;
    const float b   = *p_b;
    const float w   = *p_w;
    const float wpr = *p_pr;
    const float wil = *p_il;
    const float wol = *p_ol;
    const float wfr = *p_fr;
    const float avg = *avg_ptr;

    const int d = idxs[i];
    const float* row = gf + (size_t)(unsigned)d * N_FEAT;
    const float tf   = row[COL_TF];
    const float dl   = row[COL_DOCLEN];
    const float inl  = row[COL_INLINK];
    const float outl = row[COL_OUTLINK];
    const float pr   = row[COL_PAGERANK];

    const float numer = tf * (k1 + 1.0f);
    const float denom = tf + k1 * (1.0f - b + b * (dl / avg));
    const float bm25  = idf * (numer / denom);
    const float pscore = wpr * pr + wil * inl + wol * outl;

    // fresh_noise = jax.random.uniform(key(1), (n,)) -- threefry counter split
    const unsigned H  = (unsigned)n / 2u;
    const unsigned ui = (unsigned)i;
    const bool second = ui >= H;
    const unsigned j  = second ? (ui - H) : ui;
    unsigned r0, r1;
    threefry2x32_key01(j, j + H, r0, r1);
    const unsigned bits = second ? r1 : r0;
    const float noise = __uint_as_float((bits >> 9) | 0x3f800000u) - 1.0f;

    out[i] = w * bm25 + pscore + noise * wfr;
}

// ---------------------------------------------------------------------------
extern "C" void kernel_launch(void* const* d_in, const int* in_sizes, int n_in,
                              void* d_out, int out_size, void* d_ws, size_t ws_size,
                              hipStream_t stream) {
    const int*   idxs = (const int*)d_in[0];
    const float* gf   = (const float*)d_in[1];
    const float* k1   = (const float*)d_in[2];
    const float* b    = (const float*)d_in[3];
    const float* w    = (const float*)d_in[4];
    const float* pr   = (const float*)d_in[5];
    const float* il   = (const float*)d_in[6];
    const float* ol   = (const float*)d_in[7];
    const float* fr   = (const float*)d_in[8];
    float* out = (float*)d_out;

    const int batch  = in_sizes[0];
    const int n_docs = in_sizes[1] / N_FEAT;

    const int nblk = (n_docs + TILE - 1) / TILE;     // 245 for 1M docs
    float* partials = (float*)d_ws;                  // nblk floats
    float* avgp     = partials + nblk;               // 1 float

    // idf in f32, matching the reference arithmetic order
    const float nd  = (float)n_docs;
    const float idf = logf((nd - nd + 0.5f) / (nd + 0.5f) + 1.0f);

    col_mean_partial<<<nblk, RED_BLOCK, 0, stream>>>(gf, n_docs, partials);
    col_mean_final<<<1, RED_BLOCK, 0, stream>>>(partials, nblk, avgp, n_docs);

    const int grid = (batch + 255) / 256;
    rank_score<<<grid, 256, 0, stream>>>(idxs, gf, k1, b, w, pr, il, ol, fr,
                                         avgp, idf, out, batch);
}